// GNNCritic_54408645705761
// MI455X (gfx1250) — compile-verified
//
#include <hip/hip_runtime.h>
#include <hip/hip_bf16.h>
#include <math.h>

typedef float v2f __attribute__((ext_vector_type(2)));
typedef float v8f __attribute__((ext_vector_type(8)));

#define WAVES_PER_BLOCK 8

__global__ void __launch_bounds__(256) gnn_zero_kernel(float* __restrict__ p, int n) {
  int i = blockIdx.x * blockDim.x + threadIdx.x;
  int stride = gridDim.x * blockDim.x;
  for (; i < n; i += stride) p[i] = 0.0f;
}

// One wave32 per 16-edge tile.
// msg[e,o] = sum_i x_src[e,i]*b2[i*16+o]            (bias GEMM: 4 chained WMMA)
//          + sum_j x_src[e,j] * (h @ w2[:, j*16:+16])[e,o]   (16 x 4 chained WMMA)
// All matrix math in V_WMMA_F32_16X16X4_F32 (fp32-exact).
__global__ void __launch_bounds__(256) gnn_edge_kernel(
    const float* __restrict__ x,
    const int*   __restrict__ eidx,   // [2][E]
    const float* __restrict__ eattr,  // [E][8]
    const float* __restrict__ w1, const float* __restrict__ b1,
    const float* __restrict__ w2, const float* __restrict__ b2,
    float* __restrict__ summed, float* __restrict__ deg, int E)
{
  __shared__ float w1L[8 * 16];
  __shared__ float b1L[16];
  __shared__ float w2L[16 * 256];
  __shared__ float b2L[256];
  __shared__ __align__(16) float xsT[WAVES_PER_BLOCK][16 * 16]; // [k][e] per wave
  __shared__ __align__(16) int   dstL[WAVES_PER_BLOCK][16];

  const int tid = threadIdx.x;
  for (int i = tid; i < 8 * 16;   i += blockDim.x) w1L[i] = w1[i];
  for (int i = tid; i < 16;       i += blockDim.x) b1L[i] = b1[i];
  for (int i = tid; i < 16 * 256; i += blockDim.x) w2L[i] = w2[i];
  for (int i = tid; i < 256;      i += blockDim.x) b2L[i] = b2[i];
  __syncthreads();

  const int wave = tid >> 5;
  const int lane = tid & 31;
  const int e  = lane & 15;   // row-in-tile (A/M index)
  const int hh = lane >> 4;   // wave half
  const int col = e;          // C/D: N = lane%16
  float* xs = xsT[wave];
  int*   dl = dstL[wave];

  const v8f zero8 = {0.f, 0.f, 0.f, 0.f, 0.f, 0.f, 0.f, 0.f};

  const int ntiles  = (E + 15) >> 4;
  const int wstride = gridDim.x * WAVES_PER_BLOCK;
  for (int tile = blockIdx.x * WAVES_PER_BLOCK + wave; tile < ntiles; tile += wstride) {
    const int e0  = tile << 4;
    const int eg  = e0 + e;
    const int egc = (eg < E) ? eg : (E - 1);
    const int src = eidx[egc];
    if (hh == 0) dl[e] = eidx[E + egc];

    // gather x[src] into LDS transposed: xs[k*16 + e]; each half handles 8 features
    {
      const float4* xp = (const float4*)(x + (size_t)src * 16 + hh * 8);
      float4 xa = xp[0];
      float4 xb = xp[1];
      const int kb = hh * 8;
      xs[(kb + 0) * 16 + e] = xa.x;
      xs[(kb + 1) * 16 + e] = xa.y;
      xs[(kb + 2) * 16 + e] = xa.z;
      xs[(kb + 3) * 16 + e] = xa.w;
      xs[(kb + 4) * 16 + e] = xb.x;
      xs[(kb + 5) * 16 + e] = xb.y;
      xs[(kb + 6) * 16 + e] = xb.z;
      xs[(kb + 7) * 16 + e] = xb.w;
    }

    // h = relu(edge_attr @ w1 + b1), produced directly in WMMA f32 A-layout:
    // chain c holds K = 4c + 2*hh + {0,1} for row M = e
    const float4* ap = (const float4*)(eattr + (size_t)egc * 8);
    float4 a0 = ap[0], a1 = ap[1];
    float ea[8] = {a0.x, a0.y, a0.z, a0.w, a1.x, a1.y, a1.z, a1.w};
    v2f hA[4];
#pragma unroll
    for (int c = 0; c < 4; ++c) {
      const int k0 = 4 * c + 2 * hh;
      float s0 = b1L[k0], s1 = b1L[k0 + 1];
#pragma unroll
      for (int d = 0; d < 8; ++d) {
        s0 = fmaf(ea[d], w1L[d * 16 + k0],     s0);
        s1 = fmaf(ea[d], w1L[d * 16 + k0 + 1], s1);
      }
      hA[c].x = fmaxf(s0, 0.0f);
      hA[c].y = fmaxf(s1, 0.0f);
    }

    // make per-wave LDS tiles (xs, dl) visible to all lanes of this wave
    asm volatile("s_wait_dscnt 0" ::: "memory");

    // x_src in A layout (for bias GEMM), from the transposed LDS tile
    v2f xA[4];
#pragma unroll
    for (int c = 0; c < 4; ++c) {
      const int kk = 4 * c + 2 * hh;
      xA[c].x = xs[(kk    ) * 16 + e];
      xA[c].y = xs[(kk + 1) * 16 + e];
    }

    // acc starts as the bias term: X_src @ B2, B2[i][o] = b2[i*16+o]
    v8f accv = zero8;
#pragma unroll
    for (int c = 0; c < 4; ++c) {
      const int kk = 4 * c + 2 * hh;
      v2f bb;
      bb.x = b2L[(kk    ) * 16 + col];
      bb.y = b2L[(kk + 1) * 16 + col];
      accv = __builtin_amdgcn_wmma_f32_16x16x4_f32(
                 false, xA[c], false, bb, (short)0, accv, false, false);
    }
    float acc[8];
#pragma unroll
    for (int v = 0; v < 8; ++v) acc[v] = accv[v];

#pragma unroll 4
    for (int j = 0; j < 16; ++j) {
      v8f cc = zero8;
#pragma unroll
      for (int c = 0; c < 4; ++c) {
        const int kk = 4 * c + 2 * hh;      // B-layout mirrors A: K = 4c + 2*half + vgpr
        v2f bb;
        bb.x = w2L[(kk    ) * 256 + j * 16 + col];
        bb.y = w2L[(kk + 1) * 256 + j * 16 + col];
        cc = __builtin_amdgcn_wmma_f32_16x16x4_f32(
                 false, hA[c], false, bb, (short)0, cc, false, false);
      }
      // broadcast x_src[e, j] along o; C/D row M = v + 8*hh
      const float4* xr = (const float4*)(xs + j * 16 + hh * 8);
      float4 x0 = xr[0], x1 = xr[1];
      float xv[8] = {x0.x, x0.y, x0.z, x0.w, x1.x, x1.y, x1.z, x1.w};
#pragma unroll
      for (int v = 0; v < 8; ++v) acc[v] = fmaf(xv[v], cc[v], acc[v]);
    }

    // scatter-add messages; degrees counted by lanes 0..15
    const int4* dp = (const int4*)(dl + hh * 8);
    int4 d0 = dp[0], d1 = dp[1];
    int dn[8] = {d0.x, d0.y, d0.z, d0.w, d1.x, d1.y, d1.z, d1.w};
#pragma unroll
    for (int v = 0; v < 8; ++v) {
      if (e0 + hh * 8 + v < E) {
        atomicAdd(&summed[(size_t)dn[v] * 16 + col], acc[v]);
      }
    }
    if (hh == 0 && eg < E) atomicAdd(&deg[dl[e]], 1.0f);
  }
}

__device__ __forceinline__ float elu1(float v) { return v > 0.0f ? v : expm1f(v); }

// One 64-thread block per agent; hidden activations staged in LDS (no scratch).
__global__ void __launch_bounds__(64) gnn_head_kernel(
    const float* __restrict__ x,
    const float* __restrict__ summed, const float* __restrict__ deg,
    const int*   __restrict__ mask_idx, const float* __restrict__ action,
    const float* __restrict__ root, const float* __restrict__ bias,
    const float* __restrict__ fc1_w, const float* __restrict__ fc1_b,
    const float* __restrict__ fc2_w, const float* __restrict__ fc2_b,
    const float* __restrict__ fc3_w, const float* __restrict__ fc3_b,
    float* __restrict__ out, int nAgents)
{
  __shared__ float xmL[17];
  __shared__ float y1L[64];
  __shared__ float y2L[64];

  const int a = blockIdx.x;
  if (a >= nAgents) return;
  const int u = threadIdx.x;
  const int n = mask_idx[a];

  // xm = [aggr + x@root + bias, action]
  if (u < 16) {
    const float dinv = 1.0f / fmaxf(deg[n], 1.0f);
    const float* xn = x + (size_t)n * 16;
    float s = summed[(size_t)n * 16 + u] * dinv + bias[u];
#pragma unroll
    for (int i = 0; i < 16; ++i) s = fmaf(xn[i], root[i * 16 + u], s);
    xmL[u] = s;
  } else if (u == 16) {
    xmL[16] = action[a];
  }
  __syncthreads();

  {
    float s = fc1_b[u];
#pragma unroll
    for (int i = 0; i < 17; ++i) s = fmaf(xmL[i], fc1_w[i * 64 + u], s);
    y1L[u] = elu1(s);
  }
  __syncthreads();

  {
    float s = fc2_b[u];
#pragma unroll
    for (int i = 0; i < 64; ++i) s = fmaf(y1L[i], fc2_w[i * 64 + u], s);
    y2L[u] = elu1(s);
  }
  __syncthreads();

  if (u == 0) {
    float y = fc3_b[0];
#pragma unroll
    for (int i = 0; i < 64; ++i) y = fmaf(y2L[i], fc3_w[i], y);
    out[a] = y;
  }
}

extern "C" void kernel_launch(void* const* d_in, const int* in_sizes, int n_in,
                              void* d_out, int out_size, void* d_ws, size_t ws_size,
                              hipStream_t stream) {
  const float* x      = (const float*)d_in[0];
  const int*   eidx   = (const int*)d_in[1];
  const float* eattr  = (const float*)d_in[2];
  const int*   mask   = (const int*)d_in[3];
  const float* action = (const float*)d_in[4];
  // d_in[5] = B (unused scalar)
  const float* w1     = (const float*)d_in[6];
  const float* b1     = (const float*)d_in[7];
  const float* w2     = (const float*)d_in[8];
  const float* b2     = (const float*)d_in[9];
  const float* root   = (const float*)d_in[10];
  const float* bias   = (const float*)d_in[11];
  const float* fc1_w  = (const float*)d_in[12];
  const float* fc1_b  = (const float*)d_in[13];
  const float* fc2_w  = (const float*)d_in[14];
  const float* fc2_b  = (const float*)d_in[15];
  const float* fc3_w  = (const float*)d_in[16];
  const float* fc3_b  = (const float*)d_in[17];

  const int N = in_sizes[0] / 16;
  const int E = in_sizes[2] / 8;
  const int nAgents = in_sizes[4];

  float* summed = (float*)d_ws;            // [N][16]
  float* deg    = summed + (size_t)N * 16; // [N], contiguous after summed

  const int zn = N * 17;
  gnn_zero_kernel<<<(zn + 255) / 256, 256, 0, stream>>>(summed, zn);

  const int ntiles = (E + 15) / 16;
  int blocks = (ntiles + WAVES_PER_BLOCK - 1) / WAVES_PER_BLOCK;
  if (blocks > 1024) blocks = 1024;
  gnn_edge_kernel<<<blocks, 256, 0, stream>>>(x, eidx, eattr, w1, b1, w2, b2,
                                              summed, deg, E);

  gnn_head_kernel<<<nAgents, 64, 0, stream>>>(
      x, summed, deg, mask, action, root, bias,
      fc1_w, fc1_b, fc2_w, fc2_b, fc3_w, fc3_b, (float*)d_out, nAgents);
}